// GCN_79310866087946
// MI455X (gfx1250) — compile-verified
//
#include <hip/hip_runtime.h>

#define N_NODES 100000
#define E_EDGES 3200000
#define IN_DIM  256
#define HID     16
#define NCLS    40

typedef __attribute__((ext_vector_type(2))) float v2f;
typedef __attribute__((ext_vector_type(8))) float v8f;

// ---------------------------------------------------------------------------
// Zero the SpMM accumulators (z1 and z2 are contiguous in ws).
// ---------------------------------------------------------------------------
__global__ void zero_kernel(float* __restrict__ p, int n) {
    int i = blockIdx.x * blockDim.x + threadIdx.x;
    if (i < n) p[i] = 0.0f;
}

// ---------------------------------------------------------------------------
// XW = X[N,256] @ W1[256,16] via V_WMMA_F32_16X16X4_F32.
// One wave32 computes one 16x16 output tile (16 rows x all 16 HID cols),
// looping K in steps of 4 (64 WMMAs per tile).
//
// A (16x4 f32):  lane = {half(1b), m(4b)};  a.x = X[m][k+2*half], a.y = next K
// B (4x16 f32):  b.x = W1[k+2*half][m],     b.y = W1[k+2*half+1][m]
// C/D (16x16):   acc[v] -> row (v + 8*half), col m
// ---------------------------------------------------------------------------
__global__ void gemm1_wmma(const float* __restrict__ X,
                           const float* __restrict__ W1,
                           float* __restrict__ XW, int n_tiles) {
    int wave = (blockIdx.x * blockDim.x + threadIdx.x) >> 5;
    if (wave >= n_tiles) return;            // wave-uniform guard: EXEC all-1s inside
    int lane = threadIdx.x & 31;
    int half = lane >> 4;                   // 0 or 1
    int m    = lane & 15;
    long long rowBase = (long long)wave * 16;

    const float* arow = X + (rowBase + m) * IN_DIM;
    v8f acc = {0.f, 0.f, 0.f, 0.f, 0.f, 0.f, 0.f, 0.f};

    for (int k = 0; k < IN_DIM; k += 4) {
        int kk = k + 2 * half;
        v2f a = *(const v2f*)(arow + kk);   // contiguous pair, 8B aligned
        v2f b;
        b.x = W1[(kk + 0) * HID + m];
        b.y = W1[(kk + 1) * HID + m];
        // (neg_a, A, neg_b, B, c_mod, C, reuse_a, reuse_b)
        acc = __builtin_amdgcn_wmma_f32_16x16x4_f32(
                  false, a, false, b, (short)0, acc, false, false);
    }

    #pragma unroll
    for (int v = 0; v < 8; ++v) {
        XW[(rowBase + v + 8 * half) * HID + m] = acc[v];
    }
}

// ---------------------------------------------------------------------------
// z[dst] += val * feat[src] over 16 features.
// Thread -> (edge, feature): 16 consecutive lanes hit 16 consecutive floats of
// one destination row -> coalesced atomic bursts; [N,16] target is L2-resident.
// ---------------------------------------------------------------------------
__global__ void spmm16(const int*   __restrict__ esrc,
                       const int*   __restrict__ edst,
                       const float* __restrict__ eval,
                       const float* __restrict__ feat,
                       float* __restrict__ out) {
    unsigned tid = blockIdx.x * blockDim.x + threadIdx.x;
    if (tid >= (unsigned)E_EDGES * HID) return;
    unsigned f = tid & 15u;
    unsigned e = tid >> 4;
    int s = esrc[e];
    int d = edst[e];
    atomicAdd(out + (long long)d * HID + f,
              eval[e] * feat[(long long)s * HID + f]);
}

// ---------------------------------------------------------------------------
// In-place ReLU on h = z1.
// ---------------------------------------------------------------------------
__global__ void relu_kernel(float* __restrict__ p, int n) {
    int i = blockIdx.x * blockDim.x + threadIdx.x;
    if (i < n) p[i] = fmaxf(p[i], 0.0f);
}

// ---------------------------------------------------------------------------
// Fused: OUT[r] = log_softmax(Z2[r] @ W2).  W2 (16x40 = 2.56 KB) staged in LDS.
// One thread per row; 640 FMAs + 40-wide softmax per thread.
// ---------------------------------------------------------------------------
__global__ void out_kernel(const float* __restrict__ Z2,
                           const float* __restrict__ W2,
                           float* __restrict__ OUT) {
    __shared__ float w2s[HID * NCLS];
    for (int i = threadIdx.x; i < HID * NCLS; i += blockDim.x) w2s[i] = W2[i];
    __syncthreads();

    int r = blockIdx.x * blockDim.x + threadIdx.x;
    if (r >= N_NODES) return;

    float h[HID];
    #pragma unroll
    for (int k = 0; k < HID; ++k) h[k] = Z2[(long long)r * HID + k];

    float o[NCLS];
    #pragma unroll
    for (int c = 0; c < NCLS; ++c) {
        float acc = 0.0f;
        #pragma unroll
        for (int k = 0; k < HID; ++k) acc = fmaf(h[k], w2s[k * NCLS + c], acc);
        o[c] = acc;
    }

    float mx = o[0];
    #pragma unroll
    for (int c = 1; c < NCLS; ++c) mx = fmaxf(mx, o[c]);
    float s = 0.0f;
    #pragma unroll
    for (int c = 0; c < NCLS; ++c) s += __expf(o[c] - mx);
    float lse = mx + __logf(s);
    #pragma unroll
    for (int c = 0; c < NCLS; ++c) OUT[(long long)r * NCLS + c] = o[c] - lse;
}

// ---------------------------------------------------------------------------
// Launch: xw = X@W1 (WMMA) -> z1 = A@xw (atomics) -> relu -> z2 = A@h -> out.
// ws layout: [xw: N*16][z1: N*16][z2: N*16] floats = 19.2 MB.
// ---------------------------------------------------------------------------
extern "C" void kernel_launch(void* const* d_in, const int* in_sizes, int n_in,
                              void* d_out, int out_size, void* d_ws, size_t ws_size,
                              hipStream_t stream) {
    const float* x  = (const float*)d_in[0];
    const float* ev = (const float*)d_in[1];
    const float* w1 = (const float*)d_in[2];
    const float* w2 = (const float*)d_in[3];
    const int*   es = (const int*)d_in[4];
    const int*   ed = (const int*)d_in[5];
    float* out = (float*)d_out;

    float* xw = (float*)d_ws;
    float* z1 = xw + (size_t)N_NODES * HID;
    float* z2 = z1 + (size_t)N_NODES * HID;

    // zero z1 and z2 (contiguous) every call -> deterministic replays
    int nz = 2 * N_NODES * HID;
    zero_kernel<<<(nz + 255) / 256, 256, 0, stream>>>(z1, nz);

    // dense GEMM1 with WMMA: 6250 tiles, 8 waves per 256-thread block
    int n_tiles = N_NODES / 16;
    int blocks  = (n_tiles + 7) / 8;
    gemm1_wmma<<<blocks, 256, 0, stream>>>(x, w1, xw, n_tiles);

    // SpMM layer 1
    unsigned work = (unsigned)E_EDGES * HID;
    spmm16<<<(work + 255) / 256, 256, 0, stream>>>(es, ed, ev, xw, z1);

    // ReLU
    int nh = N_NODES * HID;
    relu_kernel<<<(nh + 255) / 256, 256, 0, stream>>>(z1, nh);

    // SpMM layer 2
    spmm16<<<(work + 255) / 256, 256, 0, stream>>>(es, ed, ev, z1, z2);

    // GEMM2 + log_softmax fused
    out_kernel<<<(N_NODES + 255) / 256, 256, 0, stream>>>(z2, w2, out);
}